// SelfDFA_3281355014354
// MI455X (gfx1250) — compile-verified
//
#include <hip/hip_runtime.h>
#include <cstddef>

// ---------------------------------------------------------------------------
// MI455X / gfx1250 wave32 WMMA implementation of SelfDFA (deformable attn TPV)
// All GEMM-shaped math uses V_WMMA_F32_16X16X4_F32 (fp32, K=4 per WMMA, chained over K in k-steps).
// One wave (32 threads) owns one 16x16 output tile.
// ---------------------------------------------------------------------------

typedef __attribute__((ext_vector_type(2))) float v2f;
typedef __attribute__((ext_vector_type(8))) float v8f;

__device__ __forceinline__ v8f wmma4(v2f a, v2f b, v8f c) {
  // D(16x16,f32) = A(16x4,f32) * B(4x16,f32) + C
  return __builtin_amdgcn_wmma_f32_16x16x4_f32(false, a, false, b, (short)0, c,
                                               false, false);
}

__device__ __forceinline__ v2f ld2(const float* p) {
  return *reinterpret_cast<const v2f*>(p);
}
__device__ __forceinline__ v2f ld2s(const float* p) {  // LDS 8B load
  return *reinterpret_cast<const v2f*>(p);
}

// ---------------------------------------------------------------------------
// Instance norm (affine=False), per channel over H*W, biased variance.
__global__ void sdfa_instnorm(const float* __restrict__ x,
                              float* __restrict__ y, int HW) {
  int c = blockIdx.x;
  const float* xc = x + (size_t)c * HW;
  float s = 0.f, s2 = 0.f;
  for (int i = threadIdx.x; i < HW; i += blockDim.x) {
    float v = xc[i]; s += v; s2 += v * v;
  }
  __shared__ float bs[256], bs2[256];
  bs[threadIdx.x] = s; bs2[threadIdx.x] = s2;
  __syncthreads();
  for (int o = 128; o > 0; o >>= 1) {
    if ((int)threadIdx.x < o) {
      bs[threadIdx.x] += bs[threadIdx.x + o];
      bs2[threadIdx.x] += bs2[threadIdx.x + o];
    }
    __syncthreads();
  }
  float mu = bs[0] / (float)HW;
  float var = bs2[0] / (float)HW - mu * mu;
  float rs = rsqrtf(var + 1e-5f);
  float* yc = y + (size_t)c * HW;
  for (int i = threadIdx.x; i < HW; i += blockDim.x) yc[i] = (xc[i] - mu) * rs;
}

// ---------------------------------------------------------------------------
// Grouped 1x1 conv: Y[g*16+m][n] = sum_k Wg[g][m][k] * X[g*16+k][n]
// grid = (G, N/16), block = 32 (one wave).
__global__ void sdfa_gconv1x1(const float* __restrict__ Wg,
                              const float* __restrict__ X,
                              float* __restrict__ Y, int N) {
  int g = blockIdx.x, nt = blockIdx.y;
  int lane = threadIdx.x, half = lane >> 4, idx = lane & 15;
  int col = nt * 16 + idx;
  v8f acc = {};
  for (int kk = 0; kk < 4; kk++) {
    int kb = 4 * kk + 2 * half;
    v2f a = ld2(&Wg[g * 256 + idx * 16 + kb]);  // contiguous pair -> b64
    v2f b;
    b.x = X[(size_t)(g * 16 + kb) * N + col];
    b.y = X[(size_t)(g * 16 + kb + 1) * N + col];
    acc = wmma4(a, b, acc);
  }
  for (int r = 0; r < 8; r++) {
    int m = half * 8 + r;
    Y[(size_t)(g * 16 + m) * N + col] = acc[r];
  }
}

// ---------------------------------------------------------------------------
// Offset net: depthwise 6x6 stride-4 conv (pad 1) on gq, exact GELU,
// 1x1 (16->2), tanh * 4, then build vgrid_scaled (faithful: ch0/(hd-1), ch1/(wd-1)).
__global__ void sdfa_offsets(const float* __restrict__ q,
                             const float* __restrict__ w1,
                             const float* __restrict__ b1,
                             const float* __restrict__ w2,
                             float* __restrict__ vgrid,
                             int H, int W, int hd, int wd) {
  int g = blockIdx.x;
  int j = threadIdx.x;
  int J = hd * wd;
  if (j >= J) return;
  int oy = j / wd, ox = j % wd;
  float acc0 = 0.f, acc1 = 0.f;
  for (int c = 0; c < 16; c++) {
    float s = b1[c];
    const float* qc = q + (size_t)(g * 16 + c) * H * W;
    for (int ky = 0; ky < 6; ky++) {
      int iy = oy * 4 - 1 + ky;
      if (iy < 0 || iy >= H) continue;
      for (int kx = 0; kx < 6; kx++) {
        int ix = ox * 4 - 1 + kx;
        if (ix < 0 || ix >= W) continue;
        s += qc[iy * W + ix] * w1[c * 36 + ky * 6 + kx];
      }
    }
    float gel = 0.5f * s * (1.f + erff(s * 0.70710678118654752f));
    acc0 += w2[c] * gel;         // row 0 of (2,16)
    acc1 += w2[16 + c] * gel;    // row 1
  }
  float vx = (float)ox + tanhf(acc0) * 4.0f;
  float vy = (float)oy + tanhf(acc1) * 4.0f;
  int dh = hd - 1 > 1 ? hd - 1 : 1;
  int dw = wd - 1 > 1 ? wd - 1 : 1;
  vgrid[(size_t)(g * J + j) * 2 + 0] = 2.f * vx / (float)dh - 1.f;
  vgrid[(size_t)(g * J + j) * 2 + 1] = 2.f * vy / (float)dw - 1.f;
}

// ---------------------------------------------------------------------------
// Bilinear grid_sample (zeros padding, align_corners=False) of grouped xn.
__global__ void sdfa_gridsample(const float* __restrict__ xn,
                                const float* __restrict__ vgrid,
                                float* __restrict__ kv,
                                int H, int W, int hd, int wd) {
  int g = blockIdx.x;
  int j = threadIdx.x;
  int J = hd * wd;
  if (j >= J) return;
  float gx = vgrid[(size_t)(g * J + j) * 2 + 0];
  float gy = vgrid[(size_t)(g * J + j) * 2 + 1];
  float x = ((gx + 1.f) * (float)W - 1.f) * 0.5f;
  float y = ((gy + 1.f) * (float)H - 1.f) * 0.5f;
  float x0 = floorf(x), y0 = floorf(y);
  float wx1 = x - x0, wy1 = y - y0, wx0 = 1.f - wx1, wy0 = 1.f - wy1;
  int ix0 = (int)x0, iy0 = (int)y0;
  for (int c = 0; c < 16; c++) {
    const float* img = xn + (size_t)(g * 16 + c) * H * W;
    float v00 = (ix0 >= 0 && ix0 < W && iy0 >= 0 && iy0 < H) ? img[iy0 * W + ix0] : 0.f;
    float v10 = (ix0 + 1 >= 0 && ix0 + 1 < W && iy0 >= 0 && iy0 < H) ? img[iy0 * W + ix0 + 1] : 0.f;
    float v01 = (ix0 >= 0 && ix0 < W && iy0 + 1 >= 0 && iy0 + 1 < H) ? img[(iy0 + 1) * W + ix0] : 0.f;
    float v11 = (ix0 + 1 >= 0 && ix0 + 1 < W && iy0 + 1 >= 0 && iy0 + 1 < H) ? img[(iy0 + 1) * W + ix0 + 1] : 0.f;
    kv[(size_t)(g * 16 + c) * J + j] =
        v00 * wx0 * wy0 + v10 * wx1 * wy0 + v01 * wx0 * wy1 + v11 * wx1 * wy1;
  }
}

// ---------------------------------------------------------------------------
// CPB relative-position-bias MLP (2->16->16->1), batched 16 kv-points per tile.
// Layer1: one 16x16x4 WMMA (K padded 2->4); layer2: 4 WMMA k-steps via LDS
// re-layout; layer3: per-lane dot + cross-half shuffle reduce.
// Loop-invariant weight fragments hoisted out of the jt loop.
// grid = (G, I), block = 32.
__global__ void sdfa_cpb_bias(const float* __restrict__ vgrid,
                              const float* __restrict__ w1,
                              const float* __restrict__ b1,
                              const float* __restrict__ w2,
                              const float* __restrict__ b2,
                              const float* __restrict__ w3,
                              const float* __restrict__ b3,
                              float* __restrict__ bias,
                              int H, int W, int hd, int wd) {
  int g = blockIdx.x;
  int i = blockIdx.y;            // query index 0..I-1
  int I = H * W, J = hd * wd;
  int lane = threadIdx.x, half = lane >> 4, idx = lane & 15;
  __shared__ float h1s[16 * 16];
  int qw = i % W, qrow = i / W;
  int dH = H - 1 > 1 ? H - 1 : 1;
  int dW = W - 1 > 1 ? W - 1 : 1;
  float gq0 = 2.f * (float)qw / (float)dH - 1.f;    // faithful: x / (H-1)
  float gq1 = 2.f * (float)qrow / (float)dW - 1.f;  // faithful: y / (W-1)
  float b3v = b3[0];
  // hoisted loop-invariant operand fragments
  v2f a1;                                   // layer-1 A (K padded 2->4)
  a1.x = half ? 0.f : w1[idx * 2 + 0];
  a1.y = half ? 0.f : w1[idx * 2 + 1];
  v2f w2a[4];                               // layer-2 A fragments
  for (int kk = 0; kk < 4; kk++) w2a[kk] = ld2(&w2[idx * 16 + 4 * kk + 2 * half]);
  float b1r[8], b2r[8], w3r[8];
  for (int r = 0; r < 8; r++) {
    int m = half * 8 + r;
    b1r[r] = b1[m]; b2r[r] = b2[m]; w3r[r] = w3[m];
  }
  for (int jt = 0; jt < J / 16; jt++) {
    int j = jt * 16 + idx;
    float p0 = gq0 - vgrid[(size_t)(g * J + j) * 2 + 0];
    float p1 = gq1 - vgrid[(size_t)(g * J + j) * 2 + 1];
    float f0 = copysignf(log1pf(fabsf(p0)), p0);
    float f1 = copysignf(log1pf(fabsf(p1)), p1);
    v2f b;
    b.x = half ? 0.f : f0;
    b.y = half ? 0.f : f1;
    v8f acc = {};
    acc = wmma4(a1, b, acc);
    for (int r = 0; r < 8; r++) {
      int m = half * 8 + r;
      h1s[m * 16 + idx] = fmaxf(acc[r] + b1r[r], 0.f);
    }
    __syncthreads();
    v8f acc2 = {};
    for (int kk = 0; kk < 4; kk++) {
      int kb = 4 * kk + 2 * half;
      v2f bb;
      bb.x = h1s[kb * 16 + idx];
      bb.y = h1s[(kb + 1) * 16 + idx];
      acc2 = wmma4(w2a[kk], bb, acc2);
    }
    float part = 0.f;
    for (int r = 0; r < 8; r++)
      part += w3r[r] * fmaxf(acc2[r] + b2r[r], 0.f);
    part += __shfl_xor(part, 16, 32);
    if (half == 0)
      bias[((size_t)g * I + i) * J + j] = part + b3v;
    __syncthreads();
  }
}

// ---------------------------------------------------------------------------
// Fused attention for one head h and one 16-row query tile:
// S = (Q*scale) K^T (WMMA) + bias; softmax over J in LDS; O = P V (WMMA).
// Q fragments hoisted; next-tile bias/K prefetched (global_prefetch_b8).
// grid = (HEADS, I/16), block = 32, dynamic LDS = 16*J floats.
__global__ void sdfa_attn(const float* __restrict__ q,
                          const float* __restrict__ kbuf,
                          const float* __restrict__ vbuf,
                          const float* __restrict__ bias,
                          float* __restrict__ attnout, int I, int J) {
  extern __shared__ float smem[];  // [16][J]
  int h = blockIdx.x, it = blockIdx.y;
  int lane = threadIdx.x, half = lane >> 4, idx = lane & 15;
  const float scale = 0.25f;  // DIM_HEAD^-0.5
  int irow = it * 16 + idx;
  // hoisted Q A-fragments (loop-invariant over jt)
  v2f qa[4];
  for (int kk = 0; kk < 4; kk++) {
    int kb = 4 * kk + 2 * half;
    qa[kk].x = q[(size_t)(h * 16 + kb) * I + irow] * scale;
    qa[kk].y = q[(size_t)(h * 16 + kb + 1) * I + irow] * scale;
  }
  for (int jt = 0; jt < J / 16; jt++) {
    int jcol = jt * 16 + idx;
    if (jt + 1 < J / 16) {  // hide next tile's latency (speculative prefetch)
      __builtin_prefetch(&bias[((size_t)h * I + irow) * J + jcol + 16], 0, 1);
      __builtin_prefetch(&kbuf[(size_t)(h * 16 + 2 * half) * J + jcol + 16], 0, 1);
    }
    v8f acc = {};
    for (int kk = 0; kk < 4; kk++) {
      int kb = 4 * kk + 2 * half;
      v2f b;
      b.x = kbuf[(size_t)(h * 16 + kb) * J + jcol];
      b.y = kbuf[(size_t)(h * 16 + kb + 1) * J + jcol];
      acc = wmma4(qa[kk], b, acc);
    }
    for (int r = 0; r < 8; r++) {
      int m = half * 8 + r;
      int gi = it * 16 + m;
      smem[m * J + jcol] = acc[r] + bias[((size_t)h * I + gi) * J + jcol];
    }
  }
  __syncthreads();
  if (lane < 16) {
    float* row = smem + lane * J;
    float mx = -3.402823466e38f;
    for (int j2 = 0; j2 < J; j2++) mx = fmaxf(mx, row[j2]);
    float sum = 0.f;
    for (int j2 = 0; j2 < J; j2++) { float e = __expf(row[j2] - mx); row[j2] = e; sum += e; }
    float inv = 1.f / sum;
    for (int j2 = 0; j2 < J; j2++) row[j2] *= inv;
  }
  __syncthreads();
  v8f acc = {};
  const float* vrow = vbuf + (size_t)(h * 16 + idx) * J;  // B[k=j][n=d]
  for (int kk = 0; kk < J / 4; kk++) {
    int kb = 4 * kk + 2 * half;
    v2f a = ld2s(&smem[idx * J + kb]);  // contiguous pair -> ds_load_b64
    v2f b = ld2(&vrow[kb]);             // contiguous pair -> global_load_b64
    acc = wmma4(a, b, acc);
  }
  for (int r = 0; r < 8; r++) {
    int m = half * 8 + r;  // query row in tile
    attnout[(size_t)(h * 16 + idx) * I + it * 16 + m] = acc[r];
  }
}

// ---------------------------------------------------------------------------
// Output projection (64x64) + bias + residual: hx = Wo*attnout + bo + x.
// grid = (4, I/16), block = 32.
__global__ void sdfa_wo(const float* __restrict__ attnout,
                        const float* __restrict__ wo,
                        const float* __restrict__ bo,
                        const float* __restrict__ xorig,
                        float* __restrict__ hx, int I) {
  int ot = blockIdx.x, nt = blockIdx.y;
  int lane = threadIdx.x, half = lane >> 4, idx = lane & 15;
  int gi = nt * 16 + idx;
  v8f acc = {};
  for (int kk = 0; kk < 16; kk++) {
    int kb = 4 * kk + 2 * half;
    v2f a = ld2(&wo[(ot * 16 + idx) * 64 + kb]);
    v2f b;
    b.x = attnout[(size_t)kb * I + gi];
    b.y = attnout[(size_t)(kb + 1) * I + gi];
    acc = wmma4(a, b, acc);
  }
  for (int r = 0; r < 8; r++) {
    int o = ot * 16 + half * 8 + r;
    hx[(size_t)o * I + gi] = acc[r] + bo[o] + xorig[(size_t)o * I + gi];
  }
}

// ---------------------------------------------------------------------------
// FFN: out = y + W2*relu(W1*x + b1) + b2 (per token), both layers WMMA.
// grid = I/16, block = 32, 16KB static LDS for the 256x16 hidden tile.
__global__ void sdfa_ffn(const float* __restrict__ xorig,
                         const float* __restrict__ y,
                         const float* __restrict__ w1,
                         const float* __restrict__ b1,
                         const float* __restrict__ w2,
                         const float* __restrict__ b2,
                         float* __restrict__ out, int I) {
  __shared__ float hid[256 * 16];
  int nt = blockIdx.x;
  int lane = threadIdx.x, half = lane >> 4, idx = lane & 15;
  int gi = nt * 16 + idx;
  for (int mt = 0; mt < 16; mt++) {
    v8f acc = {};
    for (int kk = 0; kk < 16; kk++) {
      int kb = 4 * kk + 2 * half;
      v2f a = ld2(&w1[(mt * 16 + idx) * 64 + kb]);
      v2f b;
      b.x = xorig[(size_t)kb * I + gi];
      b.y = xorig[(size_t)(kb + 1) * I + gi];
      acc = wmma4(a, b, acc);
    }
    for (int r = 0; r < 8; r++) {
      int m = mt * 16 + half * 8 + r;
      hid[m * 16 + idx] = fmaxf(acc[r] + b1[m], 0.f);
    }
  }
  __syncthreads();
  for (int ot = 0; ot < 4; ot++) {
    v8f acc = {};
    for (int kk = 0; kk < 64; kk++) {
      int kb = 4 * kk + 2 * half;
      v2f a = ld2(&w2[(ot * 16 + idx) * 256 + kb]);
      v2f b;
      b.x = hid[kb * 16 + idx];
      b.y = hid[(kb + 1) * 16 + idx];
      acc = wmma4(a, b, acc);
    }
    for (int r = 0; r < 8; r++) {
      int o = ot * 16 + half * 8 + r;
      out[(size_t)o * I + gi] = acc[r] + b2[o] + y[(size_t)o * I + gi];
    }
  }
}

// ---------------------------------------------------------------------------
// Param order assumption: d_in[0..2] = tpv_xy, tpv_xz, tpv_yz; then params
// flattened plane-major (xy, xz, yz), _plane_params insertion order:
enum { P_WQ = 0, P_WK, P_WV, P_WO, P_BO, P_OFFW1, P_OFFB1, P_OFFW2,
       P_CPBW1, P_CPBB1, P_CPBW2, P_CPBB2, P_CPBW3, P_CPBB3,
       P_FFNW1, P_FFNB1, P_FFNW2, P_FFNB2, P_COUNT };

extern "C" void kernel_launch(void* const* d_in, const int* in_sizes, int n_in,
                              void* d_out, int out_size, void* d_ws, size_t ws_size,
                              hipStream_t stream) {
  (void)in_sizes; (void)n_in; (void)out_size; (void)ws_size;
  const int Hs[3] = {64, 64, 64};
  const int Ws[3] = {64, 32, 32};
  float* ws = (float*)d_ws;
  float* out = (float*)d_out;
  size_t wsoff = 0, outoff = 0;

  for (int p = 0; p < 3; p++) {
    int H = Hs[p], W = Ws[p];
    int I = H * W, hd = H / 4, wd = W / 4, J = hd * wd;
    const float* x = (const float*)d_in[p];
    const float* prm[P_COUNT];
    for (int k = 0; k < P_COUNT; k++) prm[k] = (const float*)d_in[3 + p * P_COUNT + k];

    // workspace carve-out (floats)
    float* xn      = ws + wsoff; wsoff += (size_t)64 * I;
    float* qb      = ws + wsoff; wsoff += (size_t)64 * I;
    float* vgrid   = ws + wsoff; wsoff += (size_t)4 * J * 2;
    float* kv      = ws + wsoff; wsoff += (size_t)64 * J;
    float* kb      = ws + wsoff; wsoff += (size_t)64 * J;
    float* vb      = ws + wsoff; wsoff += (size_t)64 * J;
    float* biasb   = ws + wsoff; wsoff += (size_t)4 * I * J;
    float* attnout = ws + wsoff; wsoff += (size_t)64 * I;
    float* hx      = ws + wsoff; wsoff += (size_t)64 * I;
    float* yb      = ws + wsoff; wsoff += (size_t)64 * I;

    sdfa_instnorm<<<64, 256, 0, stream>>>(x, xn, I);
    sdfa_gconv1x1<<<dim3(4, I / 16), 32, 0, stream>>>(prm[P_WQ], xn, qb, I);
    sdfa_offsets<<<4, 256, 0, stream>>>(qb, prm[P_OFFW1], prm[P_OFFB1],
                                        prm[P_OFFW2], vgrid, H, W, hd, wd);
    sdfa_gridsample<<<4, 256, 0, stream>>>(xn, vgrid, kv, H, W, hd, wd);
    sdfa_gconv1x1<<<dim3(4, J / 16), 32, 0, stream>>>(prm[P_WK], kv, kb, J);
    sdfa_gconv1x1<<<dim3(4, J / 16), 32, 0, stream>>>(prm[P_WV], kv, vb, J);
    sdfa_cpb_bias<<<dim3(4, I), 32, 0, stream>>>(vgrid, prm[P_CPBW1], prm[P_CPBB1],
                                                 prm[P_CPBW2], prm[P_CPBB2],
                                                 prm[P_CPBW3], prm[P_CPBB3],
                                                 biasb, H, W, hd, wd);
    sdfa_attn<<<dim3(4, I / 16), 32, (size_t)16 * J * sizeof(float), stream>>>(
        qb, kb, vb, biasb, attnout, I, J);
    sdfa_wo<<<dim3(4, I / 16), 32, 0, stream>>>(attnout, prm[P_WO], prm[P_BO],
                                                x, hx, I);
    sdfa_instnorm<<<64, 256, 0, stream>>>(hx, yb, I);
    sdfa_ffn<<<I / 16, 32, 0, stream>>>(x, yb, prm[P_FFNW1], prm[P_FFNB1],
                                        prm[P_FFNW2], prm[P_FFNB2],
                                        out + outoff, I);
    outoff += (size_t)64 * I;
  }
}